// DetectionLoss_35605278884125
// MI455X (gfx1250) — compile-verified
//
#include <hip/hip_runtime.h>
#include <hip/hip_bf16.h>
#include <math.h>

#define BATCH   32
#define NCLS    8
#define HH      160
#define NANCH   (HH * HH)     /* 25600 anchors */
#define MGT     50
#define BLK     256

typedef __attribute__((ext_vector_type(2))) float v2f;
typedef __attribute__((ext_vector_type(8))) float v8f;

#define AS1_INT(p) ((__attribute__((address_space(1))) int*)(p))
#define AS3_INT(p) ((__attribute__((address_space(3))) int*)(p))

// ---------------------------------------------------------------------------
// Wave-level (32-lane) sum using V_WMMA_F32_16X16X4_F32 as a 64-way adder.
// A (16x4 f32) holds {x, 0} per lane, B = ones(4x16)  ->  D[m,n] = rowsum(m).
// Lane L holds D rows {j + 8*(L>=16)} in its 8 D VGPRs; summing them gives
// sum of rows 0..7 (lanes 0..15) / rows 8..15 (lanes 16..31); one xor-16
// shuffle add produces the full 32-lane sum in every lane.
// ---------------------------------------------------------------------------
__device__ __forceinline__ float wave_sum_32(float x) {
#if defined(__AMDGCN__) && __has_builtin(__builtin_amdgcn_wmma_f32_16x16x4_f32)
    v2f a;   a[0] = x;    a[1] = 0.0f;
    v2f one; one[0] = 1.0f; one[1] = 1.0f;
    v8f c = {0.0f, 0.0f, 0.0f, 0.0f, 0.0f, 0.0f, 0.0f, 0.0f};
    v8f d = __builtin_amdgcn_wmma_f32_16x16x4_f32(
        /*neg_a=*/false, a, /*neg_b=*/false, one,
        /*c_mod=*/(short)0, c, /*reuse_a=*/false, /*reuse_b=*/false);
    float s = d[0] + d[1] + d[2] + d[3] + d[4] + d[5] + d[6] + d[7];
    s += __shfl_xor(s, 16, 32);
    return s;
#else
    for (int m = 16; m >= 1; m >>= 1) x += __shfl_xor(x, m, 32);
    return x;
#endif
}

__device__ __forceinline__ float smooth_l1(float x) {
    float ax = fabsf(x);
    return (ax < 1.0f) ? 0.5f * x * x : ax - 0.5f;
}

// ---------------------------------------------------------------------------
// Workspace accumulators (8 floats per batch, 6 used):
//  0: sum posf (npos)          1: sum negf (nneg)
//  2: sum ce_tgt * posf        3: sum ce_bg * negf
//  4: sum ce_bg (all anchors)  5: sum smoothL1 * posf
// ---------------------------------------------------------------------------
__global__ void det_zero_acc(float* __restrict__ acc) {
    acc[threadIdx.x] = 0.0f;   // 256 floats
}

__global__ void det_loss_main(const float* __restrict__ cls,
                              const float* __restrict__ reg,
                              const float* __restrict__ gt_boxes,
                              const int*   __restrict__ gt_labels,
                              const int*   __restrict__ num_boxes,
                              float*       __restrict__ acc) {
    __shared__ float s_box[MGT * 4];
    __shared__ int   s_lab[MGT];
    __shared__ int   s_nb;
    __shared__ float s_red[6 * BLK];

    const int b   = blockIdx.y;
    const int tid = threadIdx.x;
    const int n   = blockIdx.x * BLK + tid;   // N = 100 * 256 exactly, no tail

    // ---- stage GT data for this batch into LDS (async path when available) --
    const float* gb = gt_boxes + (size_t)b * MGT * 4;
    const int*   gl = gt_labels + (size_t)b * MGT;
#if defined(__AMDGCN__) && __has_builtin(__builtin_amdgcn_global_load_async_to_lds_b32) && __has_builtin(__builtin_amdgcn_s_wait_asynccnt)
    if (tid < MGT * 4)
        __builtin_amdgcn_global_load_async_to_lds_b32(
            AS1_INT(gb + tid), AS3_INT(s_box + tid), 0, 0);
    if (tid < MGT)
        __builtin_amdgcn_global_load_async_to_lds_b32(
            AS1_INT(gl + tid), AS3_INT(s_lab + tid), 0, 0);
    if (tid == 0) s_nb = num_boxes[b];
    __builtin_amdgcn_s_wait_asynccnt(0);
    __syncthreads();
#else
    if (tid < MGT * 4) s_box[tid] = gb[tid];
    if (tid < MGT)     s_lab[tid] = gl[tid];
    if (tid == 0)      s_nb = num_boxes[b];
    __syncthreads();
#endif
    const int nb = s_nb;

    // ---- analytic anchor (all 32x32 boxes, stride-4 grid, row-major) -------
    const float acx = (float)(n % HH) * 4.0f + 2.0f;
    const float acy = (float)(n / HH) * 4.0f + 2.0f;

    // ---- load regression (coalesced planes) and decode ---------------------
    const float* regb = reg + (size_t)b * 4 * NANCH + n;
    const float r0 = regb[0];
    const float r1 = regb[(size_t)NANCH];
    const float r2 = regb[(size_t)2 * NANCH];
    const float r3 = regb[(size_t)3 * NANCH];

    // prefetch the class-logit planes we will need after the IoU scan
    const float* clsb = cls + (size_t)b * NCLS * NANCH + n;
    __builtin_prefetch(clsb, 0, 1);
    __builtin_prefetch(clsb + (size_t)4 * NANCH, 0, 1);

    const float tx = r0 * 2.0f - 1.0f;
    const float ty = r1 * 2.0f - 1.0f;
    const float cx = acx + tx * 8.0f;          // aw/4 = 8
    const float cy = acy + ty * 8.0f;
    const float w  = 32.0f * expf(r2);
    const float h  = 32.0f * expf(r3);
    const float dx1 = cx - 0.5f * w, dy1 = cy - 0.5f * h;
    const float dx2 = cx + 0.5f * w, dy2 = cy + 0.5f * h;
    const float area1 = w * h;

    // ---- IoU scan over GT boxes (first-max semantics like jnp.argmax) ------
    float best = -2.0f;
    int   bidx = 0;
    for (int j = 0; j < MGT; ++j) {
        const float g0 = s_box[j * 4 + 0], g1 = s_box[j * 4 + 1];
        const float g2 = s_box[j * 4 + 2], g3 = s_box[j * 4 + 3];
        const float lx = fmaxf(dx1, g0), ly = fmaxf(dy1, g1);
        const float rx = fminf(dx2, g2), ry = fminf(dy2, g3);
        const float iw = fmaxf(rx - lx, 0.0f), ih = fmaxf(ry - ly, 0.0f);
        const float inter = iw * ih;
        const float a2 = (g2 - g0) * (g3 - g1);
        const float iou = inter / fmaxf(area1 + a2 - inter, 1e-8f);
        const float v = (j < nb) ? iou : -1.0f;
        if (v > best) { best = v; bidx = j; }
    }
    const bool has = nb > 0;
    const bool pos = (best >= 0.25f) && has;
    const bool neg = (best < 0.10f) && has;

    // ---- classification: 8-way log-softmax ---------------------------------
    float x[NCLS];
#pragma unroll
    for (int k = 0; k < NCLS; ++k) x[k] = clsb[(size_t)k * NANCH];
    float m = x[0];
#pragma unroll
    for (int k = 1; k < NCLS; ++k) m = fmaxf(m, x[k]);
    float se = 0.0f;
#pragma unroll
    for (int k = 0; k < NCLS; ++k) se += expf(x[k] - m);
    const float lse = m + logf(se);
    const int   tgt = s_lab[bidx];
    const float ce_tgt = lse - x[tgt];
    const float ce_bg  = lse - x[0];

    // ---- regression smooth-L1 vs matched GT --------------------------------
    const float g0 = s_box[bidx * 4 + 0], g1 = s_box[bidx * 4 + 1];
    const float g2 = s_box[bidx * 4 + 2], g3 = s_box[bidx * 4 + 3];
    const float gw = g2 - g0, gh = g3 - g1;
    const float gcx = g0 + 0.5f * gw, gcy = g1 + 0.5f * gh;
    const float ttx = ((gcx - acx) * 0.125f + 1.0f) * 0.5f;   // (4/aw)=0.125
    const float tty = ((gcy - acy) * 0.125f + 1.0f) * 0.5f;
    const float ttw = logf(fmaxf(gw, 1e-6f) * (1.0f / 32.0f));
    const float tth = logf(fmaxf(gh, 1e-6f) * (1.0f / 32.0f));
    const float sl = smooth_l1(r0 - ttx) + smooth_l1(r1 - tty) +
                     smooth_l1(r2 - ttw) + smooth_l1(r3 - tth);

    // ---- block reduction: 6 quantities via LDS + WMMA wave-sum -------------
    s_red[0 * BLK + tid] = pos ? 1.0f : 0.0f;
    s_red[1 * BLK + tid] = neg ? 1.0f : 0.0f;
    s_red[2 * BLK + tid] = pos ? ce_tgt : 0.0f;
    s_red[3 * BLK + tid] = neg ? ce_bg : 0.0f;
    s_red[4 * BLK + tid] = ce_bg;
    s_red[5 * BLK + tid] = pos ? sl : 0.0f;
    __syncthreads();

    if (tid < 32) {
#pragma unroll
        for (int q = 0; q < 6; ++q) {
            float s = 0.0f;
#pragma unroll
            for (int i = 0; i < BLK / 32; ++i) s += s_red[q * BLK + tid + i * 32];
            const float tot = wave_sum_32(s);   // v_wmma_f32_16x16x4_f32
            if (tid == 0) atomicAdd(&acc[b * 8 + q], tot);
        }
    }
}

// ---------------------------------------------------------------------------
// Finalize: one wave, lane b handles batch b, WMMA-reduce across batches.
// ---------------------------------------------------------------------------
__global__ void det_loss_final(const float* __restrict__ acc,
                               const int*   __restrict__ num_boxes,
                               float*       __restrict__ out) {
    const int b = threadIdx.x;                 // 0..31 == BATCH
    const float npos = acc[b * 8 + 0];
    const float nneg = acc[b * 8 + 1];
    const float sct  = acc[b * 8 + 2];
    const float scb  = acc[b * 8 + 3];
    const float sca  = acc[b * 8 + 4];
    const float ssl  = acc[b * 8 + 5];
    const bool  has  = num_boxes[b] > 0;

    const float cls_pos = (npos > 0.0f) ? sct / fmaxf(npos, 1.0f) : 0.0f;
    const float cls_neg = (nneg > 0.0f) ? scb / fmaxf(nneg, 1.0f) : 0.0f;
    const float cls_l   = has ? (cls_pos + cls_neg) : sca * (1.0f / (float)NANCH);
    const float reg_l   = (npos > 0.0f) ? ssl / fmaxf(npos * 4.0f, 1.0f) : 0.0f;

    const float cls_sum = wave_sum_32(cls_l);
    const float reg_sum = wave_sum_32(reg_l);
    const float tp      = wave_sum_32(npos);

    if (b == 0) {
        const float cls_final = cls_sum * (1.0f / (float)BATCH);
        const float reg_final = reg_sum / fmaxf(tp, 1.0f);
        out[0] = cls_final + reg_final;
        out[1] = cls_final;
        out[2] = reg_final;
        out[3] = tp;
    }
}

extern "C" void kernel_launch(void* const* d_in, const int* in_sizes, int n_in,
                              void* d_out, int out_size, void* d_ws, size_t ws_size,
                              hipStream_t stream) {
    const float* cls = (const float*)d_in[0];   // [32,8,1,160,160] f32
    const float* reg = (const float*)d_in[1];   // [32,4,1,160,160] f32
    // d_in[2] = anchors: analytic, not needed
    const float* gtb = (const float*)d_in[3];   // [32,50,4] f32
    const int*   gtl = (const int*)d_in[4];     // [32,50] i32
    const int*   nbx = (const int*)d_in[5];     // [32] i32
    float* out = (float*)d_out;                 // 4 f32 scalars
    float* acc = (float*)d_ws;                  // 256 f32 accumulators

    det_zero_acc<<<1, 256, 0, stream>>>(acc);
    det_loss_main<<<dim3(NANCH / BLK, BATCH), BLK, 0, stream>>>(cls, reg, gtb, gtl, nbx, acc);
    det_loss_final<<<1, 32, 0, stream>>>(acc, nbx, out);
}